// WFDM_NoWts_46273977647934
// MI455X (gfx1250) — compile-verified
//
#include <hip/hip_runtime.h>
#include <stdint.h>

// ---------------------------------------------------------------------------
// 2x2 box-average (sum * 0.5) + 2x2 nearest upsample, fp32 (16,256,128,128).
// Memory-bound: 512 MiB HBM traffic -> ~23 us floor at 23.3 TB/s.
// Load path: CDNA5 async global->LDS copy engine (ASYNCcnt), double-buffered,
// two quads (2 rows x 4 cols each) per pipeline stage. All address math is
// strength-reduced to constant-delta pointer bumps; row offset (512 B) folds
// into the vmem instruction's immediate offset.
// ---------------------------------------------------------------------------

__device__ __forceinline__ void async_ld_b128(unsigned lds_off, const float* g) {
    asm volatile("global_load_async_to_lds_b128 %0, %1, off"
                 :: "v"(lds_off), "v"(g) : "memory");
}

// ---------------- fast path: requires nq % (2*T) == 0, T % 32 == 0 ----------
__global__ __launch_bounds__(256) void avg2x2_up_fast(
    const float* __restrict__ x, float* __restrict__ o,
    int iters, unsigned dq /* = 8*T elements: distance between the 2 quads */) {
    __shared__ __align__(16) float sb[2][256][16];   // 32 KB double buffer

    const unsigned tid = threadIdx.x;
    const unsigned q   = blockIdx.x * blockDim.x + tid;

    unsigned lds[2];
    lds[0] = (unsigned)(uintptr_t)(void*)&sb[0][tid][0];
    lds[1] = (unsigned)(uintptr_t)(void*)&sb[1][tid][0];

    // q = slice*2048 + rowpair*32 + col4  ->  base = 8q - 4(q&31).
    // (q&31) is iteration-invariant because the stage stride is 2*dq = 16*T
    // elements and T % 32 == 0, so bases advance by a constant delta.
    const unsigned base0 = 8u * q - 4u * (q & 31u);
    const size_t   dstg  = 2 * (size_t)dq;           // elements per stage

    const float* pinA  = x + base0;
    const float* pinB  = pinA + dq;
    float*       poutA = o + base0;
    float*       poutB = poutA + dq;

    // Prologue: stage 0 -> buffer 0 (4 async b128 copies; +128 elem = 512 B
    // row offset is an immediate).
    async_ld_b128(lds[0],       pinA);
    async_ld_b128(lds[0] + 16u, pinA + 128);
    async_ld_b128(lds[0] + 32u, pinB);
    async_ld_b128(lds[0] + 48u, pinB + 128);
    pinA += dstg;  pinB += dstg;

    int buf = 0;
    for (int i = 0; i < iters; ++i) {
        if (i + 1 < iters) {
            const unsigned l = lds[buf ^ 1];
            async_ld_b128(l,       pinA);
            async_ld_b128(l + 16u, pinA + 128);
            async_ld_b128(l + 32u, pinB);
            async_ld_b128(l + 48u, pinB + 128);
            pinA += dstg;  pinB += dstg;
            // 8 outstanding; loads complete in order, so <=4 means the
            // current buffer's 4 copies have landed in LDS.
            asm volatile("s_wait_asynccnt 4" ::: "memory");
        } else {
            asm volatile("s_wait_asynccnt 0" ::: "memory");
        }

        const float4 a0 = *(const float4*)&sb[buf][tid][0];
        const float4 a1 = *(const float4*)&sb[buf][tid][4];
        const float4 b0 = *(const float4*)&sb[buf][tid][8];
        const float4 b1 = *(const float4*)&sb[buf][tid][12];

        float va0 = (a0.x + a0.y + a1.x + a1.y) * 0.5f;
        float va1 = (a0.z + a0.w + a1.z + a1.w) * 0.5f;
        float vb0 = (b0.x + b0.y + b1.x + b1.y) * 0.5f;
        float vb1 = (b0.z + b0.w + b1.z + b1.w) * 0.5f;
        const float4 wa = make_float4(va0, va0, va1, va1);
        const float4 wb = make_float4(vb0, vb0, vb1, vb1);

        *(float4*)(poutA)       = wa;   // row 2*rp
        *(float4*)(poutA + 128) = wa;   // row 2*rp+1 (512 B immediate)
        *(float4*)(poutB)       = wb;
        *(float4*)(poutB + 128) = wb;
        poutA += dstg;  poutB += dstg;
        buf ^= 1;
    }
}

// ---------------- generic fallback (any nq), unroll-1 pipeline --------------
__global__ __launch_bounds__(256) void avg2x2_up_generic(
    const float* __restrict__ x, float* __restrict__ o, unsigned nq) {
    __shared__ __align__(16) float sb[2][256][8];

    const unsigned tid    = threadIdx.x;
    const unsigned stride = gridDim.x * blockDim.x;

    unsigned lds[2];
    lds[0] = (unsigned)(uintptr_t)(void*)&sb[0][tid][0];
    lds[1] = (unsigned)(uintptr_t)(void*)&sb[1][tid][0];

    unsigned q   = blockIdx.x * blockDim.x + tid;
    int      buf = 0;

    if (q < nq) {
        unsigned base = 8u * q - 4u * (q & 31u);
        async_ld_b128(lds[0],       x + base);
        async_ld_b128(lds[0] + 16u, x + base + 128u);
    }
    while (q < nq) {
        unsigned nextq = q + stride;
        if (nextq < nq) {
            unsigned nb = 8u * nextq - 4u * (nextq & 31u);
            async_ld_b128(lds[buf ^ 1],       x + nb);
            async_ld_b128(lds[buf ^ 1] + 16u, x + nb + 128u);
            asm volatile("s_wait_asynccnt 2" ::: "memory");
        } else {
            asm volatile("s_wait_asynccnt 0" ::: "memory");
        }

        const float4 r0 = *(const float4*)&sb[buf][tid][0];
        const float4 r1 = *(const float4*)&sb[buf][tid][4];
        float v0 = (r0.x + r0.y + r1.x + r1.y) * 0.5f;
        float v1 = (r0.z + r0.w + r1.z + r1.w) * 0.5f;
        float4 w = make_float4(v0, v0, v1, v1);

        unsigned base = 8u * q - 4u * (q & 31u);
        *(float4*)(o + base)        = w;
        *(float4*)(o + base + 128u) = w;

        q = nextq;
        buf ^= 1;
    }
}

extern "C" void kernel_launch(void* const* d_in, const int* in_sizes, int n_in,
                              void* d_out, int out_size, void* d_ws, size_t ws_size,
                              hipStream_t stream) {
    (void)in_sizes; (void)n_in; (void)d_ws; (void)ws_size;
    const float* x = (const float*)d_in[0];
    float*       o = (float*)d_out;

    // Output shape (16,256,128,128): one quad = 2 rows x 4 cols = 8 elements.
    const unsigned nq = (unsigned)(out_size / 8);

    const int threads = 256;                       // 8 wave32 waves per block
    const int blocks  = 1024;                      // 262,144 threads = 8192 waves
    const unsigned T  = (unsigned)threads * (unsigned)blocks;

    if (nq % (2u * T) == 0u) {
        // nq = 2^23 for the reference shape -> exactly 16 uniform stages.
        const int      iters = (int)(nq / (2u * T));
        const unsigned dq    = 8u * T;             // elements between quad A and B
        avg2x2_up_fast<<<blocks, threads, 0, stream>>>(x, o, iters, dq);
    } else {
        unsigned need = (nq + threads - 1) / threads;
        int      gb   = (need < 2048u) ? (int)need : 2048;
        if (gb < 1) gb = 1;
        avg2x2_up_generic<<<gb, threads, 0, stream>>>(x, o, nq);
    }
}